// GATLayer_22385369547441
// MI455X (gfx1250) — compile-verified
//
#include <hip/hip_runtime.h>
#include <hip/hip_bf16.h>
#include <math.h>

#define NN        50000      // nodes
#define EE        800000     // edges
#define IN_F      256
#define OUT_F     32
#define NH        8          // heads
#define HF        (NH*OUT_F) // 256 = concat width
#define NEG_SLOPE 0.2f
#define EPS       1e-8f

#define RT        5          // row tiles (of 16) per block: 50000 = 80 * 625
#define RPB       (RT*16)    // 80 rows per block

typedef __attribute__((ext_vector_type(2))) float v2f;
typedef __attribute__((ext_vector_type(8))) float v8f;

// ---------------- init: zero out, -inf max buffer, zero sum buffer ----------
__global__ void k_init(float* __restrict__ out, float* __restrict__ m,
                       float* __restrict__ ssum) {
    long i = (long)blockIdx.x * blockDim.x + threadIdx.x;
    if (i < (long)NN * HF) out[i] = 0.0f;
    if (i < (long)NN * NH) { m[i] = -INFINITY; ssum[i] = 0.0f; }
}

// ---------------- GEMM: h[n, head*32+f] = sum_k x[n,k] * W[head,k,f] --------
// One block = 80 output rows (5 row tiles, A in LDS), 8 waves -> 8 col tiles.
// Register blocking: each wave loads one B fragment per k-step and reuses it
// across 5 WMMAs (5 accumulators), cutting B-side L2 traffic 5x.
__global__ __launch_bounds__(256) void k_gemm(const float* __restrict__ x,
                                              const float* __restrict__ W,
                                              float* __restrict__ hout) {
    __shared__ float As[RPB * IN_F]; // 80KB A tile
    const int r0 = blockIdx.x * RPB;

    // cooperative load: 80 rows x 64 float4/row = 5120 float4, 20 per thread
    const float4* xg = (const float4*)(x + (long)r0 * IN_F);
    float4* As4 = (float4*)As;
#pragma unroll
    for (int i = 0; i < 20; ++i)
        As4[threadIdx.x + i * 256] = xg[threadIdx.x + i * 256];
    __syncthreads();

    const int w    = threadIdx.x >> 5;   // wave id: column tile
    const int lane = threadIdx.x & 31;
    const int c0   = w * 16;             // first output column of this wave
    const int n    = lane & 15;          // N (and M for A) index within tile
    const int half = lane >> 4;          // 0: K=0,1   1: K=2,3
    const int head = c0 >> 5;
    const int fcol = (c0 & 31) + n;
    // B[k][c0+n] = W[head*IN_F*OUT_F + k*OUT_F + fcol]
    const float* Bp = W + head * (IN_F * OUT_F) + fcol;

    v8f acc[RT];
#pragma unroll
    for (int rt = 0; rt < RT; ++rt)
        acc[rt] = (v8f){0.f,0.f,0.f,0.f,0.f,0.f,0.f,0.f};

#pragma unroll 4
    for (int kk = 0; kk < IN_F; kk += 4) {
        const int ka = kk + half * 2;
        v2f b;
        b.x = Bp[(long)ka * OUT_F];                  // B[K=ka,   N=n]
        b.y = Bp[(long)(ka + 1) * OUT_F];            // B[K=ka+1, N=n]
#pragma unroll
        for (int rt = 0; rt < RT; ++rt) {
            v2f a = *(const v2f*)(&As[(rt * 16 + n) * IN_F + ka]);
            acc[rt] = __builtin_amdgcn_wmma_f32_16x16x4_f32(
                /*neg_a=*/false, a, /*neg_b=*/false, b,
                /*c_mod=*/(short)0, acc[rt],
                /*reuse_a=*/false, /*reuse_b=*/false);
        }
    }

    // D layout: VGPR v -> row M = v + 8*half, col = n
#pragma unroll
    for (int rt = 0; rt < RT; ++rt) {
        float* op = hout + (long)(r0 + rt * 16) * HF + c0 + n;
#pragma unroll
        for (int v = 0; v < 8; ++v)
            op[(long)(v + half * 8) * HF] = acc[rt][v];
    }
}

// ---------------- per-node attention logits --------------------------------
__global__ void k_scores(const float* __restrict__ h,
                         const float* __restrict__ a_src,
                         const float* __restrict__ a_dst,
                         float* __restrict__ ssrc, float* __restrict__ sdst) {
    int t = blockIdx.x * blockDim.x + threadIdx.x;
    if (t >= NN * NH) return;
    const int nnode = t >> 3, hh = t & 7;
    const float* hp = h + (long)nnode * HF + hh * OUT_F;
    const float* as = a_src + hh * OUT_F;
    const float* ad = a_dst + hh * OUT_F;
    float s1 = 0.f, s2 = 0.f;
#pragma unroll
    for (int f = 0; f < OUT_F; ++f) {
        float v = hp[f];
        s1 += v * as[f];
        s2 += v * ad[f];
    }
    ssrc[t] = s1;
    sdst[t] = s2;
}

// float atomic max via signed-max / unsigned-min bit trick
__device__ __forceinline__ void atomicMaxF(float* addr, float val) {
    if (val >= 0.0f) atomicMax((int*)addr, __float_as_int(val));
    else             atomicMin((unsigned int*)addr, __float_as_uint(val));
}

// ---------------- edge pass 1: LeakyReLU score + segment max ---------------
__global__ void k_edge_max(const float* __restrict__ ssrc,
                           const float* __restrict__ sdst,
                           const int* __restrict__ ei,
                           float* __restrict__ ea, float* __restrict__ m) {
    int t = blockIdx.x * blockDim.x + threadIdx.x;   // t < E*H
    const int e = t >> 3, hh = t & 7;
    const int src = ei[e], dst = ei[EE + e];
    float v = ssrc[(long)src * NH + hh] + sdst[(long)dst * NH + hh];
    v = v > 0.0f ? v : NEG_SLOPE * v;
    ea[t] = v;
    atomicMaxF(&m[(long)dst * NH + hh], v);
}

// ---------------- edge pass 2: exp(score - max) + segment sum --------------
__global__ void k_edge_expsum(const int* __restrict__ ei,
                              const float* __restrict__ m,
                              float* __restrict__ ea,
                              float* __restrict__ ssum) {
    int t = blockIdx.x * blockDim.x + threadIdx.x;   // t < E*H
    const int e = t >> 3, hh = t & 7;
    const int dst = ei[EE + e];
    float v = __expf(ea[t] - m[(long)dst * NH + hh]);
    ea[t] = v;
    atomicAdd(&ssum[(long)dst * NH + hh], v);
}

// ---------------- edge pass 3: normalized weighted scatter-add -------------
// One wave per edge: 32 lanes x 8 heads cover the 256 concat features.
__global__ __launch_bounds__(256) void k_aggregate(const float* __restrict__ h,
                                                   const float* __restrict__ ea,
                                                   const float* __restrict__ ssum,
                                                   const int* __restrict__ ei,
                                                   float* __restrict__ out) {
    const int wv   = threadIdx.x >> 5;
    const int lane = threadIdx.x & 31;
    const long e = (long)blockIdx.x * 8 + wv;
    const int src = ei[e], dst = ei[EE + e];
    float wl = 0.0f;
    if (lane < NH)
        wl = ea[e * NH + lane] / (ssum[(long)dst * NH + lane] + EPS);
    const float* hs = h + (long)src * HF;
    float* od = out + (long)dst * HF;
#pragma unroll
    for (int j = 0; j < NH; ++j) {
        float wj = __shfl(wl, j, 32);
        int c = j * 32 + lane;
        atomicAdd(&od[c], hs[c] * wj);
    }
}

extern "C" void kernel_launch(void* const* d_in, const int* in_sizes, int n_in,
                              void* d_out, int out_size, void* d_ws, size_t ws_size,
                              hipStream_t stream) {
    (void)in_sizes; (void)n_in; (void)out_size; (void)ws_size;
    const float* x     = (const float*)d_in[0];
    const int*   ei    = (const int*)  d_in[1];
    const float* W     = (const float*)d_in[2];
    const float* a_src = (const float*)d_in[3];
    const float* a_dst = (const float*)d_in[4];
    float* out = (float*)d_out;

    // workspace carve-up (floats)
    float* ws   = (float*)d_ws;
    float* h    = ws;                        // N*256 = 12.8M
    float* ssrc = h    + (long)NN * HF;      // N*8
    float* sdst = ssrc + (long)NN * NH;      // N*8
    float* m    = sdst + (long)NN * NH;      // N*8
    float* ssum = m    + (long)NN * NH;      // N*8
    float* ea   = ssum + (long)NN * NH;      // E*8 = 6.4M

    // 1) init accumulators/output
    k_init<<<(NN * HF + 255) / 256, 256, 0, stream>>>(out, m, ssum);
    // 2) WMMA GEMM: h = x @ W_concat   (50000/80 = 625 register-blocked tiles)
    k_gemm<<<NN / RPB, 256, 0, stream>>>(x, W, h);
    // 3) per-node logits
    k_scores<<<(NN * NH + 255) / 256, 256, 0, stream>>>(h, a_src, a_dst, ssrc, sdst);
    // 4) edge max
    k_edge_max<<<(EE * NH) / 256, 256, 0, stream>>>(ssrc, sdst, ei, ea, m);
    // 5) edge exp + sum
    k_edge_expsum<<<(EE * NH) / 256, 256, 0, stream>>>(ei, m, ea, ssum);
    // 6) weighted scatter aggregation, one wave per edge
    k_aggregate<<<EE / 8, 256, 0, stream>>>(h, ea, ssum, ei, out);
}